// GraphPropagationNetwork_15006615733042
// MI455X (gfx1250) — compile-verified
//
#include <hip/hip_runtime.h>

typedef __attribute__((ext_vector_type(2))) float v2f;
typedef __attribute__((ext_vector_type(8))) float v8f;

#define NS   2048
#define NQ   6144
#define NTOT 8192
#define IND  512
#define DM   256
#define NC   64
#define KNN  10
#define ADJW (NTOT / 32)   // 256 words per row

// workspace byte offsets
#define OFF_E    0u
#define OFF_SQ   (8u * 1024u * 1024u)                    // 8 MB for E
#define OFF_TOPI (OFF_SQ + 32768u)                       // 32 KB for sq
#define OFF_ADJ  (OFF_TOPI + (unsigned)(NTOT * KNN * 4)) // 320 KB for topi
#define OFF_L0   (OFF_ADJ + (unsigned)(NTOT * ADJW * 4)) // 8 MB adj bitset
#define OFF_L1   (OFF_L0 + (unsigned)(NTOT * NC * 4))    // 2 MB labels each

#define BIGF 3.402823466e38f

// ---------------------------------------------------------------------------
// E = [support; query] @ W  via V_WMMA_F32_16X16X4_F32
// grid 2048 blocks (512 row-strips x 4 col-groups), block 128 (4 waves);
// each wave computes one 16x16 output tile over K=512.
// ---------------------------------------------------------------------------
__global__ void embed_kernel(const float* __restrict__ support,
                             const float* __restrict__ query,
                             const float* __restrict__ W,
                             float* __restrict__ E) {
  int bm   = blockIdx.x >> 2;
  int bn   = blockIdx.x & 3;
  int wave = threadIdx.x >> 5;
  int lane = threadIdx.x & 31;
  int M0 = bm * 16;
  int N0 = bn * 64 + wave * 16;
  int nl   = lane & 15;
  int koff = (lane >> 4) * 2;       // A: lanes 0-15 hold K 0..1, lanes 16-31 hold K 2..3
  int row  = M0 + nl;
  int n    = N0 + nl;
  const float* xr = (row < NS) ? (support + (size_t)row * IND)
                               : (query + (size_t)(row - NS) * IND);
  v8f acc = {};
  for (int k = 0; k < IND; k += 4) {
    v2f a; a.x = xr[k + koff];                  a.y = xr[k + koff + 1];
    v2f b; b.x = W[(size_t)(k + koff) * DM + n]; b.y = W[(size_t)(k + koff + 1) * DM + n];
    acc = __builtin_amdgcn_wmma_f32_16x16x4_f32(false, a, false, b,
                                                (short)0, acc, false, false);
  }
  int mh = (lane >> 4) * 8;  // D: VGPR r -> M = r (lanes 0-15), M = 8+r (lanes 16-31)
#pragma unroll
  for (int r = 0; r < 8; ++r)
    E[(size_t)(M0 + mh + r) * DM + N0 + nl] = acc[r];
}

// per-row squared norms
__global__ void sq_kernel(const float* __restrict__ E, float* __restrict__ sq) {
  int i = blockIdx.x * blockDim.x + threadIdx.x;
  if (i < NTOT) {
    const float* p = E + (size_t)i * DM;
    float s = 0.f;
    for (int c = 0; c < DM; ++c) s += p[c] * p[c];
    sq[i] = s;
  }
}

// ---------------------------------------------------------------------------
// Gram + fused top-10: 512 blocks (16-row strips) x 512 threads (16 waves).
// Per 256-col tile: each wave does a 16x16 WMMA Gram tile -> d2 into LDS,
// then one thread per wave streams 256 candidates into its private top-10.
// ---------------------------------------------------------------------------
__global__ void gram_topk_kernel(const float* __restrict__ E,
                                 const float* __restrict__ sq,
                                 int* __restrict__ topi) {
  __shared__ float Es[16 * DM];     // 16 KB strip embeddings
  __shared__ float d2buf[16 * 256]; // 16 KB distance tile
  __shared__ float sqs[16];

  int tid  = threadIdx.x;
  int wave = tid >> 5;
  int lane = tid & 31;
  int M0 = blockIdx.x * 16;

  for (int idx = tid; idx < 16 * DM; idx += 512) {
    int r = idx >> 8, c = idx & 255;
    Es[idx] = E[(size_t)(M0 + r) * DM + c];
  }
  if (tid < 16) sqs[tid] = sq[M0 + tid];
  __syncthreads();

  float td[KNN]; int ti[KNN];
#pragma unroll
  for (int m = 0; m < KNN; ++m) { td[m] = BIGF; ti[m] = 0; }

  int nl   = lane & 15;
  int koff = (lane >> 4) * 2;
  int mh   = (lane >> 4) * 8;

  for (int cb = 0; cb < NTOT / 256; ++cb) {
    int C0 = cb * 256;
    int j  = C0 + wave * 16 + nl;
    const float* Ea = Es + nl * DM + koff;
    const float* Ej = E + (size_t)j * DM + koff;
    v8f acc = {};
    for (int k = 0; k < DM; k += 4) {
      v2f a; a.x = Ea[k]; a.y = Ea[k + 1];
      v2f b; b.x = Ej[k]; b.y = Ej[k + 1];
      acc = __builtin_amdgcn_wmma_f32_16x16x4_f32(false, a, false, b,
                                                  (short)0, acc, false, false);
    }
    float sqj = sq[j];
#pragma unroll
    for (int r = 0; r < 8; ++r) {
      int M = mh + r;
      float d2 = sqs[M] + sqj - 2.0f * acc[r];
      d2buf[M * 256 + wave * 16 + nl] = d2 < 0.f ? 0.f : d2;
    }
    __syncthreads();
    if (lane == 0) {               // wave w's lane 0 owns strip row w
      int rrow = wave;
      for (int c = 0; c < 256; ++c) {
        float d = d2buf[rrow * 256 + c];
        if (d < td[KNN - 1]) {     // strict < keeps earlier index on ties (JAX top_k)
          int p = KNN - 1;
          while (p > 0 && d < td[p - 1]) {
            td[p] = td[p - 1]; ti[p] = ti[p - 1]; --p;
          }
          td[p] = d; ti[p] = C0 + c;
        }
      }
    }
    __syncthreads();
  }
  if (lane == 0) {
    int i = M0 + wave;
#pragma unroll
    for (int m = 0; m < KNN; ++m) topi[i * KNN + m] = ti[m];
  }
}

__global__ void zero_u32(unsigned* __restrict__ p, int n) {
  int i = blockIdx.x * blockDim.x + threadIdx.x;
  if (i < n) p[i] = 0u;
}

// symmetric adjacency bitset from top-10 lists
__global__ void scatter_kernel(const int* __restrict__ topi, unsigned* __restrict__ adj) {
  int e = blockIdx.x * blockDim.x + threadIdx.x;
  if (e < NTOT * KNN) {
    int i = e / KNN;
    int j = topi[e];
    atomicOr(&adj[(size_t)i * ADJW + (j >> 5)], 1u << (j & 31));
    atomicOr(&adj[(size_t)j * ADJW + (i >> 5)], 1u << (i & 31));
  }
}

__global__ void labels_init_kernel(const int* __restrict__ slab, float* __restrict__ L) {
  int idx = blockIdx.x * blockDim.x + threadIdx.x;
  if (idx < NTOT * NC) {
    int i = idx >> 6, c = idx & 63;
    L[idx] = (i < NS && slab[i] == c) ? 1.f : 0.f;
  }
}

// one block per row, one thread per class; bit-scan the adjacency row
__global__ void prop_kernel(const unsigned* __restrict__ adj,
                            const float* __restrict__ src,
                            float* __restrict__ dst) {
  int row = blockIdx.x;
  int c   = threadIdx.x;  // 0..63
  const unsigned* ar = adj + (size_t)row * ADJW;
  float acc = 0.f;
  int deg = 0;
  for (int w = 0; w < ADJW; ++w) {
    unsigned word = ar[w];
    deg += __popc(word);
    while (word) {
      int b = __ffs(word) - 1;
      word &= word - 1;
      acc += src[(size_t)(w * 32 + b) * NC + c];
    }
  }
  dst[(size_t)row * NC + c] = acc / (float)(deg > 0 ? deg : 1);
}

// argmax -> one-hot for the query rows
__global__ void output_kernel(const float* __restrict__ L, float* __restrict__ out) {
  int q = blockIdx.x * blockDim.x + threadIdx.x;
  if (q < NQ) {
    const float* row = L + (size_t)(NS + q) * NC;
    int best = 0; float bv = row[0];
    for (int c = 1; c < NC; ++c) {
      float v = row[c];
      if (v > bv) { bv = v; best = c; }  // strict > == first-max (JAX argmax)
    }
    float* o = out + (size_t)q * NC;
    for (int c = 0; c < NC; ++c) o[c] = (c == best) ? 1.f : 0.f;
  }
}

extern "C" void kernel_launch(void* const* d_in, const int* in_sizes, int n_in,
                              void* d_out, int out_size, void* d_ws, size_t ws_size,
                              hipStream_t stream) {
  const float* support = (const float*)d_in[0];
  const float* query   = (const float*)d_in[1];
  const int*   slab    = (const int*)d_in[2];
  const float* W       = (const float*)d_in[3];
  float* out = (float*)d_out;

  char* ws = (char*)d_ws;
  float*    E    = (float*)(ws + OFF_E);
  float*    sq   = (float*)(ws + OFF_SQ);
  int*      topi = (int*)(ws + OFF_TOPI);
  unsigned* adj  = (unsigned*)(ws + OFF_ADJ);
  float*    L0   = (float*)(ws + OFF_L0);
  float*    L1   = (float*)(ws + OFF_L1);

  embed_kernel<<<2048, 128, 0, stream>>>(support, query, W, E);
  sq_kernel<<<(NTOT + 255) / 256, 256, 0, stream>>>(E, sq);
  gram_topk_kernel<<<NTOT / 16, 512, 0, stream>>>(E, sq, topi);
  zero_u32<<<(NTOT * ADJW + 255) / 256, 256, 0, stream>>>(adj, NTOT * ADJW);
  scatter_kernel<<<(NTOT * KNN + 255) / 256, 256, 0, stream>>>(topi, adj);
  labels_init_kernel<<<(NTOT * NC + 255) / 256, 256, 0, stream>>>(slab, L0);
  prop_kernel<<<NTOT, NC, 0, stream>>>(adj, L0, L1);
  prop_kernel<<<NTOT, NC, 0, stream>>>(adj, L1, L0);
  prop_kernel<<<NTOT, NC, 0, stream>>>(adj, L0, L1);
  output_kernel<<<(NQ + 255) / 256, 256, 0, stream>>>(L1, out);
}